// SessionPrototypeAllocator_24446953849144
// MI455X (gfx1250) — compile-verified
//
#include <hip/hip_runtime.h>
#include <math.h>

#define B_   512
#define T_   512
#define D_   1024
#define NF_  256
#define DE_  256
#define DH_  512
#define K_   64
#define PD_  1024
#define IN_DIM (D_ + DE_)   // 1280

typedef __attribute__((ext_vector_type(2))) float v2f;
typedef __attribute__((ext_vector_type(8))) float v8f;

// ---------------------------------------------------------------------------
// Kernel 1: masked mean over feat (the 256MB streaming read) + last-hidden
// gather. One block per batch row; thread f owns feature column f (coalesced:
// 256 threads read 1KB contiguous per timestep).
// ---------------------------------------------------------------------------
__global__ __launch_bounds__(256) void pool_kernel(
    const float* __restrict__ hidden, const float* __restrict__ feat,
    const int* __restrict__ lens, float* __restrict__ pooled,
    float* __restrict__ fmean)
{
    const int b   = blockIdx.x;
    const int tid = threadIdx.x;
    int len = lens[b];
    len = len < 1 ? 1 : (len > T_ ? T_ : len);

    const float* fp = feat + (long)b * T_ * NF_ + tid;
    float s = 0.f;
    for (int t = 0; t < len; ++t) s += fp[(long)t * NF_];
    fmean[b * NF_ + tid] = s / (float)len;

    // pooling='last'
    const float* hp = hidden + (long)b * T_ * D_ + (long)(len - 1) * D_;
    float* pp = pooled + (long)b * IN_DIM;
#pragma unroll
    for (int i = 0; i < 4; ++i) pp[tid + 256 * i] = hp[tid + 256 * i];
}

// ---------------------------------------------------------------------------
// FP32 WMMA 16x16 tile: C += A(16x4) x B(4x16), K-loop in steps of 4.
// Lane layout per CDNA5 ISA 7.12.2:
//   A 16x4 : lane L -> row m=L&15;  VGPR0/1 hold K = kb, kb+1 with kb=(L>>4)*2
//   B 4x16 : lane L -> col n=L&15;  VGPR0/1 hold K = kb, kb+1
//   C 16x16: VGPR v -> row v + 8*(L>>4), col L&15
// ---------------------------------------------------------------------------
__device__ __forceinline__ v8f wmma_gemm_tile(
    const float* __restrict__ A, int lda,
    const float* __restrict__ Bm, int ldb,
    int K, int m0, int n0, int lane)
{
    const int mr = lane & 15;
    const int kb = (lane >> 4) << 1;
    const float* ap = A  + (long)(m0 + mr) * lda + kb;
    const float* bp = Bm + (long)kb * ldb + (n0 + mr);
    v8f c = {0.f, 0.f, 0.f, 0.f, 0.f, 0.f, 0.f, 0.f};
    for (int k = 0; k < K; k += 4) {
        v2f a, b;
        a.x = ap[0];
        a.y = ap[1];
        b.x = bp[0];
        b.y = bp[ldb];
        c = __builtin_amdgcn_wmma_f32_16x16x4_f32(
                false, a, false, b, (short)0, c, false, false);
        ap += 4;
        bp += 4 * (long)ldb;
    }
    return c;
}

// ---------------------------------------------------------------------------
// Kernel 2: f_emb = f_mean @ Wf + bf  -> pooled[:, 1024:1280]
// tiles = (512/16)*(256/16) = 512 waves -> 64 blocks x 8 waves
// ---------------------------------------------------------------------------
__global__ __launch_bounds__(256) void femb_kernel(
    const float* __restrict__ fmean, const float* __restrict__ Wf,
    const float* __restrict__ bf, float* __restrict__ pooled)
{
    const int wave = blockIdx.x * (blockDim.x >> 5) + (threadIdx.x >> 5);
    const int lane = threadIdx.x & 31;
    const int ntn  = DE_ / 16;             // 16
    const int mo   = wave / ntn, no = wave % ntn;

    v8f c = wmma_gemm_tile(fmean, NF_, Wf, DE_, NF_, mo * 16, no * 16, lane);

    const int col   = no * 16 + (lane & 15);
    const int rbase = mo * 16 + ((lane >> 4) << 3);
    const float bias = bf[col];
#pragma unroll
    for (int v = 0; v < 8; ++v)
        pooled[(long)(rbase + v) * IN_DIM + D_ + col] = c[v] + bias;
}

// ---------------------------------------------------------------------------
// Kernel 3: h = gelu_exact(pooled @ W1 + b1)
// tiles = 32*32 = 1024 waves -> 128 blocks
// ---------------------------------------------------------------------------
__global__ __launch_bounds__(256) void mlp1_kernel(
    const float* __restrict__ pooled, const float* __restrict__ W1,
    const float* __restrict__ b1, float* __restrict__ hbuf)
{
    const int wave = blockIdx.x * (blockDim.x >> 5) + (threadIdx.x >> 5);
    const int lane = threadIdx.x & 31;
    const int ntn  = DH_ / 16;             // 32
    const int mo   = wave / ntn, no = wave % ntn;

    v8f c = wmma_gemm_tile(pooled, IN_DIM, W1, DH_, IN_DIM, mo * 16, no * 16, lane);

    const int col   = no * 16 + (lane & 15);
    const int rbase = mo * 16 + ((lane >> 4) << 3);
    const float bias = b1[col];
#pragma unroll
    for (int v = 0; v < 8; ++v) {
        float x = c[v] + bias;
        float g = 0.5f * x * (1.0f + erff(x * 0.70710678118654752f)); // exact GELU
        hbuf[(long)(rbase + v) * DH_ + col] = g;
    }
}

// ---------------------------------------------------------------------------
// Kernel 4: logits = h @ W2 + b2   (TEMP == 1.0)
// tiles = 32*4 = 128 waves -> 16 blocks
// ---------------------------------------------------------------------------
__global__ __launch_bounds__(256) void mlp2_kernel(
    const float* __restrict__ hbuf, const float* __restrict__ W2,
    const float* __restrict__ b2, float* __restrict__ logits)
{
    const int wave = blockIdx.x * (blockDim.x >> 5) + (threadIdx.x >> 5);
    const int lane = threadIdx.x & 31;
    const int ntn  = K_ / 16;              // 4
    const int mo   = wave / ntn, no = wave % ntn;

    v8f c = wmma_gemm_tile(hbuf, DH_, W2, K_, DH_, mo * 16, no * 16, lane);

    const int col   = no * 16 + (lane & 15);
    const int rbase = mo * 16 + ((lane >> 4) << 3);
    const float bias = b2[col];
#pragma unroll
    for (int v = 0; v < 8; ++v)
        logits[(long)(rbase + v) * K_ + col] = c[v] + bias;
}

// ---------------------------------------------------------------------------
// Kernel 5: per-row top-4 + softmax gating. One thread per batch row.
// ---------------------------------------------------------------------------
__global__ __launch_bounds__(256) void topk_kernel(
    const float* __restrict__ logits, float* __restrict__ weights,
    int* __restrict__ top_idx, float* __restrict__ top_w)
{
    const int b = blockIdx.x * blockDim.x + threadIdx.x;
    if (b >= B_) return;

    const float* lr = logits + (long)b * K_;
    float tv[4] = {-INFINITY, -INFINITY, -INFINITY, -INFINITY};
    int   ti[4] = {0, 0, 0, 0};
    for (int k = 0; k < K_; ++k) {
        float v = lr[k];
        int pos = 4;
#pragma unroll
        for (int j = 3; j >= 0; --j) { if (v > tv[j]) pos = j; }
        if (pos < 4) {
            for (int j = 3; j > pos; --j) { tv[j] = tv[j - 1]; ti[j] = ti[j - 1]; }
            tv[pos] = v; ti[pos] = k;
        }
    }
    // softmax over the 4 kept values
    float m = tv[0];
    float e[4], s = 0.f;
#pragma unroll
    for (int j = 0; j < 4; ++j) { e[j] = __expf(tv[j] - m); s += e[j]; }
    float inv = 1.0f / s;

    float* wr = weights + (long)b * K_;
    for (int k = 0; k < K_; ++k) wr[k] = 0.f;
#pragma unroll
    for (int j = 0; j < 4; ++j) {
        float w = e[j] * inv;
        wr[ti[j]] = w;
        top_idx[b * 4 + j] = ti[j];
        top_w[b * 4 + j]   = w;
    }
}

// ---------------------------------------------------------------------------
// Kernel 6: context = weights @ proto, exploiting 4-sparse rows:
// context[b] = sum_j w_j * proto[idx_j].  float4-coalesced over PD=1024.
// ---------------------------------------------------------------------------
__global__ __launch_bounds__(256) void context_kernel(
    const int* __restrict__ top_idx, const float* __restrict__ top_w,
    const float* __restrict__ proto, float* __restrict__ ctx)
{
    const int b   = blockIdx.x;
    const int tid = threadIdx.x;            // one float4 per thread (1024/4=256)
    const float4* p4 = (const float4*)proto;
    float4 acc = make_float4(0.f, 0.f, 0.f, 0.f);
#pragma unroll
    for (int j = 0; j < 4; ++j) {
        const float w   = top_w[b * 4 + j];
        const int   idx = top_idx[b * 4 + j];
        float4 pv = p4[(long)idx * (PD_ / 4) + tid];
        acc.x += w * pv.x; acc.y += w * pv.y;
        acc.z += w * pv.z; acc.w += w * pv.w;
    }
    ((float4*)ctx)[(long)b * (PD_ / 4) + tid] = acc;
}

// ---------------------------------------------------------------------------
extern "C" void kernel_launch(void* const* d_in, const int* in_sizes, int n_in,
                              void* d_out, int out_size, void* d_ws, size_t ws_size,
                              hipStream_t stream)
{
    const float* hidden = (const float*)d_in[0];
    const float* feat   = (const float*)d_in[1];
    const int*   lens   = (const int*)  d_in[2];
    const float* Wf     = (const float*)d_in[3];
    const float* bf     = (const float*)d_in[4];
    const float* W1     = (const float*)d_in[5];
    const float* b1     = (const float*)d_in[6];
    const float* W2     = (const float*)d_in[7];
    const float* b2     = (const float*)d_in[8];
    const float* proto  = (const float*)d_in[9];

    float* out_w      = (float*)d_out;                 // [B,K]
    float* out_logits = out_w + B_ * K_;               // [B,K]
    float* out_ctx    = out_logits + B_ * K_;          // [B,PD]

    char* ws = (char*)d_ws;
    float* pooled = (float*)ws;  ws += (size_t)B_ * IN_DIM * sizeof(float);
    float* fmean  = (float*)ws;  ws += (size_t)B_ * NF_ * sizeof(float);
    float* hbuf   = (float*)ws;  ws += (size_t)B_ * DH_ * sizeof(float);
    int*   tidx   = (int*)ws;    ws += (size_t)B_ * 4 * sizeof(int);
    float* tw     = (float*)ws;

    pool_kernel   <<<B_,  256, 0, stream>>>(hidden, feat, lens, pooled, fmean);
    femb_kernel   <<<64,  256, 0, stream>>>(fmean, Wf, bf, pooled);
    mlp1_kernel   <<<128, 256, 0, stream>>>(pooled, W1, b1, hbuf);
    mlp2_kernel   <<<16,  256, 0, stream>>>(hbuf, W2, b2, out_logits);
    topk_kernel   <<<2,   256, 0, stream>>>(out_logits, out_w, tidx, tw);
    context_kernel<<<B_,  256, 0, stream>>>(tidx, tw, proto, out_ctx);
}